// FPS_24850680775244
// MI455X (gfx1250) — compile-verified
//
#include <hip/hip_runtime.h>
#include <cfloat>

// FPS: B=64 clouds, M=16384 pts, K=4096 samples. One workgroup per cloud,
// entire cloud resident in LDS (192KB of the 320KB/WGP), distances in VGPRs.

#define B_CLOUDS 64
#define M_PTS    16384
#define K_SAMP   4096
#define THREADS  1024
#define PTS_PER_THREAD (M_PTS / THREADS)      // 16
#define CLOUD_BYTES (M_PTS * 3 * 4)           // 196608
#define SMEM_BYTES  (CLOUD_BYTES + 512)

typedef int v4i __attribute__((vector_size(4 * sizeof(int))));

__global__ __launch_bounds__(THREADS) void fps_kernel(const float* __restrict__ pos,
                                                      float* __restrict__ out)
{
    extern __shared__ __align__(16) char smem[];
    float* sp    = (float*)smem;                        // [3*M] xyz AoS
    float* wredD = (float*)(smem + CLOUD_BYTES);        // [32] per-wave max
    int*   wredI = (int*)  (smem + CLOUD_BYTES + 128);  // [32] per-wave argmax
    float* bc    = (float*)(smem + CLOUD_BYTES + 256);  // [4] broadcast xyz

    const int tid = threadIdx.x;
    const int b   = blockIdx.x;
    const float* cloud = pos + (size_t)b * M_PTS * 3;

    // ---- Fill LDS with this cloud (gfx1250 async-to-LDS when available) ----
    constexpr int CHUNKS = CLOUD_BYTES / 16;            // 12288 x 16B
#if defined(__gfx1250__) && __has_builtin(__builtin_amdgcn_global_load_async_to_lds_b128)
    {
        typedef __attribute__((address_space(1))) char gchar;
        typedef __attribute__((address_space(3))) char lchar;
        gchar* gbase = (gchar*)cloud;
        lchar* lbase = (lchar*)smem;
        for (int j = tid; j < CHUNKS; j += THREADS) {
            __builtin_amdgcn_global_load_async_to_lds_b128(
                (__attribute__((address_space(1))) v4i*)(gbase + 16 * j),
                (__attribute__((address_space(3))) v4i*)(lbase + 16 * j),
                0, 0);
        }
    }
#if __has_builtin(__builtin_amdgcn_s_wait_asynccnt)
    __builtin_amdgcn_s_wait_asynccnt(0);
#else
    asm volatile("s_wait_asynccnt 0" ::: "memory");
#endif
#else
    {
        const float4* g4 = (const float4*)cloud;
        float4*       s4 = (float4*)smem;
        for (int j = tid; j < CHUNKS; j += THREADS) s4[j] = g4[j];
    }
#endif
    __syncthreads();

    // ---- init: first sample is local index 0; dist starts at FLT_MAX so the
    // first min-update against point 0 produces dist0 exactly like reference.
    float cx = sp[0], cy = sp[1], cz = sp[2];
    float dist[PTS_PER_THREAD];
#pragma unroll
    for (int t = 0; t < PTS_PER_THREAD; ++t) dist[t] = FLT_MAX;

    if (tid == 0) out[(size_t)b * K_SAMP] = (float)((size_t)b * M_PTS);

    for (int k = 1; k < K_SAMP; ++k) {
        float bestD = -1.0f;
        int   bestI = 0;
#pragma unroll
        for (int t = 0; t < PTS_PER_THREAD; ++t) {
            const int i = tid + (t << 10);
            const float x = sp[3 * i + 0];
            const float y = sp[3 * i + 1];
            const float z = sp[3 * i + 2];
            const float dx = x - cx, dy = y - cy, dz = z - cz;
            const float nd = fmaf(dz, dz, fmaf(dy, dy, dx * dx));
            float d = dist[t];
            d = fminf(d, nd);
            dist[t] = d;
            if (d > bestD) { bestD = d; bestI = i; }   // within-thread: first max wins
        }
        // wave32 argmax reduce, tie -> smaller index (matches jnp.argmax)
#pragma unroll
        for (int off = 16; off > 0; off >>= 1) {
            const float oD = __shfl_down(bestD, (unsigned)off, 32);
            const int   oI = __shfl_down(bestI, (unsigned)off, 32);
            if (oD > bestD || (oD == bestD && oI < bestI)) { bestD = oD; bestI = oI; }
        }
        if ((tid & 31) == 0) { wredD[tid >> 5] = bestD; wredI[tid >> 5] = bestI; }
        __syncthreads();
        if (tid < 32) {
            bestD = wredD[tid];
            bestI = wredI[tid];
#pragma unroll
            for (int off = 16; off > 0; off >>= 1) {
                const float oD = __shfl_down(bestD, (unsigned)off, 32);
                const int   oI = __shfl_down(bestI, (unsigned)off, 32);
                if (oD > bestD || (oD == bestD && oI < bestI)) { bestD = oD; bestI = oI; }
            }
            if (tid == 0) {
                bc[0] = sp[3 * bestI + 0];
                bc[1] = sp[3 * bestI + 1];
                bc[2] = sp[3 * bestI + 2];
                out[(size_t)b * K_SAMP + k] = (float)((size_t)b * M_PTS + bestI);
            }
        }
        __syncthreads();
        cx = bc[0]; cy = bc[1]; cz = bc[2];
    }
}

extern "C" void kernel_launch(void* const* d_in, const int* in_sizes, int n_in,
                              void* d_out, int out_size, void* d_ws, size_t ws_size,
                              hipStream_t stream) {
    (void)in_sizes; (void)n_in; (void)d_ws; (void)ws_size; (void)out_size;
    const float* pos = (const float*)d_in[0];   // [B*M, 3] float32; d_in[1] (batch) unused: uniform
    float* out = (float*)d_out;                 // [B*K] global indices (as float)

    // Allow >64KB dynamic LDS (host-side attribute; deterministic, not a stream op).
    (void)hipFuncSetAttribute((const void*)fps_kernel,
                              hipFuncAttributeMaxDynamicSharedMemorySize, SMEM_BYTES);

    fps_kernel<<<B_CLOUDS, THREADS, SMEM_BYTES, stream>>>(pos, out);
}